// LSTMNNMF_19636590477583
// MI455X (gfx1250) — compile-verified
//
#include <hip/hip_runtime.h>
#include <math.h>

// Problem constants (match reference)
#define N_SENS   5000
#define T_STEPS  20000
#define RANK     60
#define FOURR    240
#define MAXLAG   288
#define BATCH    (T_STEPS - MAXLAG)   // 19712 = 16*1232
#define NEDGE    80000

typedef __attribute__((ext_vector_type(2))) float v2f;
typedef __attribute__((ext_vector_type(8))) float v8f;

// D(16x16,f32) += A(16x4,f32) * B(4x16,f32)
// A layout: lanes 0-15 -> M=0..15, vgpr0/1 = K {0,1} (lanes<16) or {2,3} (lanes>=16)
// B layout: lanes 0-15 -> N=0..15, vgpr0/1 = K {0,1} / {2,3}
__device__ __forceinline__ v8f wmma4(v2f a, v2f b, v8f c) {
    return __builtin_amdgcn_wmma_f32_16x16x4_f32(
        /*neg_a=*/false, a, /*neg_b=*/false, b,
        /*c_mod=*/(short)0, c, /*reuse_a=*/false, /*reuse_b=*/false);
}

__device__ __forceinline__ void block_reduce_add(double v, double* dst) {
    __shared__ double red[256];
    int tid = threadIdx.x;
    red[tid] = v;
    __syncthreads();
    for (int s = 128; s > 0; s >>= 1) {
        if (tid < s) red[tid] += red[tid + s];
        __syncthreads();
    }
    if (tid == 0) atomicAdd(dst, red[0]);
}

// ---------------- main masked GEMM residual: sum over mask of (W@X.T - gt)^2 --------
// One wave computes a 16(M) x 80(T) tile: 5 WMMA accumulators, A reused 5x per k-step.
// gt/mask are streamed once -> prefetch before the K-loop, non-temporal loads after.
__global__ void k_main_resid(const float* __restrict__ W, const float* __restrict__ X,
                             const float* __restrict__ gt, const unsigned char* __restrict__ mask,
                             double* __restrict__ acc) {
    const int CT = T_STEPS / 80;            // 250 col tiles (exact)
    const int MT = (N_SENS + 15) / 16;      // 313 row tiles
    int wid = blockIdx.x * 8 + (threadIdx.x >> 5);
    double local = 0.0;
    if (wid < MT * CT) {                    // wave-uniform guard (EXEC all-1s inside)
        int mt = wid / CT, ct = wid % CT;
        int rowbase = mt * 16, colbase = ct * 80;
        int lane = threadIdx.x & 31;
        int half = lane >> 4, lm = lane & 15;

        int arow = rowbase + lm; if (arow >= N_SENS) arow = N_SENS - 1; // clamp keeps EXEC full
        const float* Wr = W + (long)arow * RANK;
        const float* Xp0 = X + (long)(colbase + lm) * RANK;

        // Prefetch this tile's gt/mask rows so the streaming epilogue overlaps the GEMM.
        {
            long pbase = (long)arow * T_STEPS + colbase;
            const char* gp = (const char*)(gt + pbase);
            __builtin_prefetch(gp + half * 160, 0, 1);        // 320B/row of gt
            __builtin_prefetch(gp + half * 160 + 128, 0, 1);
            const char* mp = (const char*)(mask + pbase);
            __builtin_prefetch(mp + half * 64, 0, 1);         // 80B/row of mask
        }

        v8f c0 = {0,0,0,0,0,0,0,0}, c1 = {0,0,0,0,0,0,0,0}, c2 = {0,0,0,0,0,0,0,0};
        v8f c3 = {0,0,0,0,0,0,0,0}, c4 = {0,0,0,0,0,0,0,0};
        #pragma unroll
        for (int k0 = 0; k0 < RANK; k0 += 4) {
            int k = k0 + 2 * half;
            v2f a;  a.x = Wr[k]; a.y = Wr[k + 1];
            v2f b0; b0.x = Xp0[k];                 b0.y = Xp0[k + 1];
            v2f b1; b1.x = Xp0[16L * RANK + k];    b1.y = Xp0[16L * RANK + k + 1];
            v2f b2; b2.x = Xp0[32L * RANK + k];    b2.y = Xp0[32L * RANK + k + 1];
            v2f b3; b3.x = Xp0[48L * RANK + k];    b3.y = Xp0[48L * RANK + k + 1];
            v2f b4; b4.x = Xp0[64L * RANK + k];    b4.y = Xp0[64L * RANK + k + 1];
            c0 = wmma4(a, b0, c0);
            c1 = wmma4(a, b1, c1);
            c2 = wmma4(a, b2, c2);
            c3 = wmma4(a, b3, c3);
            c4 = wmma4(a, b4, c4);
        }
        v8f cs[5] = {c0, c1, c2, c3, c4};
        #pragma unroll
        for (int j = 0; j < 5; ++j) {
            #pragma unroll
            for (int v = 0; v < 8; ++v) {
                int r = rowbase + v + 8 * half;
                if (r < N_SENS) {
                    long idx = (long)r * T_STEPS + colbase + 16 * j + lm;
                    unsigned char m = __builtin_nontemporal_load(&mask[idx]);
                    float g         = __builtin_nontemporal_load(&gt[idx]);
                    float e = cs[j][v] - g;
                    if (m) local += (double)e * e;
                }
            }
        }
    }
    block_reduce_add(local, acc + 0);
}

// ---------------- generic row-major WMMA GEMM: C[M,Nn] = A[M,K] @ B[K,Nn] ----------
__global__ void k_gemm_store(const float* __restrict__ A, const float* __restrict__ B,
                             float* __restrict__ C, int M, int K, int Nn) {
    int mtiles = (M + 15) >> 4, ntiles = (Nn + 15) >> 4;
    int wid = blockIdx.x * 8 + (threadIdx.x >> 5);
    if (wid >= mtiles * ntiles) return;     // wave-uniform
    int mt = wid / ntiles, nt = wid % ntiles;
    int rowbase = mt * 16, colbase = nt * 16;
    int lane = threadIdx.x & 31, half = lane >> 4, lm = lane & 15;
    int arow = rowbase + lm; if (arow >= M) arow = M - 1;
    int bcol = colbase + lm; if (bcol >= Nn) bcol = Nn - 1;
    v8f c = {0,0,0,0,0,0,0,0};
    for (int k0 = 0; k0 < K; k0 += 4) {
        int k = k0 + 2 * half;
        v2f a; a.x = A[(long)arow * K + k];       a.y = A[(long)arow * K + k + 1];
        v2f b; b.x = B[(long)k * Nn + bcol];      b.y = B[(long)(k + 1) * Nn + bcol];
        c = wmma4(a, b, c);
    }
    for (int v = 0; v < 8; ++v) {
        int r  = rowbase + v + 8 * half;
        int cc = colbase + lm;
        if (r < M && cc < Nn) C[(long)r * Nn + cc] = c[v];
    }
}

// ---------------- elementwise LSTM gates (keras order i,f,g,o) ---------------------
__global__ void k_lstm_gate(const float* __restrict__ XK, const float* __restrict__ Zrec,
                            const float* __restrict__ bias,
                            float* __restrict__ H, float* __restrict__ Cst,
                            int rowOff, int first) {
    int idx = blockIdx.x * blockDim.x + threadIdx.x;
    if (idx >= BATCH * RANK) return;
    int t = idx / RANK, j = idx - t * RANK;
    long zb = (long)(t + rowOff) * FOURR;
    float zi = XK[zb + j]        + bias[j];
    float zf = XK[zb + 60 + j]   + bias[60 + j];
    float zg = XK[zb + 120 + j]  + bias[120 + j];
    float zo = XK[zb + 180 + j]  + bias[180 + j];
    if (!first) {
        long rb = (long)t * FOURR;
        zi += Zrec[rb + j]; zf += Zrec[rb + 60 + j];
        zg += Zrec[rb + 120 + j]; zo += Zrec[rb + 180 + j];
    }
    float gi = 1.f / (1.f + expf(-zi));
    float gf = 1.f / (1.f + expf(-zf));
    float gg = tanhf(zg);
    float go = 1.f / (1.f + expf(-zo));
    float c  = first ? (gi * gg) : (gf * Cst[idx] + gi * gg);
    Cst[idx] = c;
    H[idx]   = go * tanhf(c);
}

// ---------------- dense layer + squared error vs X[t+288] (fused WMMA) -------------
__global__ void k_dense_resid(const float* __restrict__ H, const float* __restrict__ Dw,
                              const float* __restrict__ Db, const float* __restrict__ X,
                              double* __restrict__ acc) {
    const int mtiles = BATCH / 16;   // 1232
    const int ntiles = 4;            // cols 0..63, guard >=60
    int wid = blockIdx.x * 8 + (threadIdx.x >> 5);
    double local = 0.0;
    if (wid < mtiles * ntiles) {
        int mt = wid / ntiles, nt = wid % ntiles;
        int rowbase = mt * 16, colbase = nt * 16;
        int lane = threadIdx.x & 31, half = lane >> 4, lm = lane & 15;
        int arow = rowbase + lm;
        int bcol = colbase + lm; if (bcol >= RANK) bcol = RANK - 1;
        v8f c = {0,0,0,0,0,0,0,0};
        for (int k0 = 0; k0 < RANK; k0 += 4) {
            int k = k0 + 2 * half;
            v2f a; a.x = H[(long)arow * RANK + k];  a.y = H[(long)arow * RANK + k + 1];
            v2f b; b.x = Dw[(long)k * RANK + bcol]; b.y = Dw[(long)(k + 1) * RANK + bcol];
            c = wmma4(a, b, c);
        }
        for (int v = 0; v < 8; ++v) {
            int r  = rowbase + v + 8 * half;
            int cc = colbase + lm;
            if (cc < RANK) {
                float p = c[v] + Db[cc];
                float e = X[(long)(r + MAXLAG) * RANK + cc] - p;
                local += (double)e * e;
            }
        }
    }
    block_reduce_add(local, acc + 4);
}

// ---------------- small terms ------------------------------------------------------
__global__ void k_sumsq(const float* __restrict__ A, long n, double* __restrict__ acc, int slot) {
    long idx = (long)blockIdx.x * blockDim.x + threadIdx.x;
    long stride = (long)gridDim.x * blockDim.x;
    double local = 0.0;
    for (long i = idx; i < n; i += stride) { float v = A[i]; local += (double)v * v; }
    block_reduce_add(local, acc + slot);
}

__global__ void k_graph(const float* __restrict__ W, const int* __restrict__ src,
                        const int* __restrict__ dst, double* __restrict__ acc) {
    int e = blockIdx.x * blockDim.x + threadIdx.x;
    double local = 0.0;
    if (e < NEDGE) {
        const float* ws = W + (long)src[e] * RANK;
        const float* wd = W + (long)dst[e] * RANK;
        for (int r = 0; r < RANK; ++r) { float d = ws[r] - wd[r]; local += (double)d * d; }
    }
    block_reduce_add(local, acc + 3);
}

__global__ void k_init(double* acc) { if (threadIdx.x < 5) acc[threadIdx.x] = 0.0; }

__global__ void k_final(const double* __restrict__ acc, float* __restrict__ out) {
    if (threadIdx.x == 0) {
        double tre = acc[0];
        double wf  = acc[1] * (100.0 * 0.1);   // LAMBDA_W * ETA
        double xf  = acc[2] * (100.0 * 0.1);   // LAMBDA_X * ETA
        double wn  = acc[3] * 100.0;           // LAMBDA_W
        double xn  = acc[4] * 100.0;           // LAMBDA_X
        out[0] = (float)(tre + wf + xf + xn + wn);
        out[1] = (float)tre;
        out[2] = (float)wf;
        out[3] = (float)xf;
        out[4] = (float)xn;
    }
}

extern "C" void kernel_launch(void* const* d_in, const int* in_sizes, int n_in,
                              void* d_out, int out_size, void* d_ws, size_t ws_size,
                              hipStream_t stream) {
    const float*         W    = (const float*)d_in[0];
    const float*         X    = (const float*)d_in[1];
    const float*         gt   = (const float*)d_in[2];
    const unsigned char* mask = (const unsigned char*)d_in[3];  // jax bool = 1 byte
    const int*           esrc = (const int*)d_in[4];
    const int*           edst = (const int*)d_in[5];
    const float*         lk   = (const float*)d_in[6];
    const float*         lrk  = (const float*)d_in[7];
    const float*         lb   = (const float*)d_in[8];
    const float*         dw   = (const float*)d_in[9];
    const float*         db   = (const float*)d_in[10];

    // workspace layout (~48 MB): [5 f64 acc][XK 20000x240][H B x60][C B x60][Zrec B x240]
    char*   ws  = (char*)d_ws;
    double* acc = (double*)ws;
    float*  XK  = (float*)(ws + 64);
    float*  H   = XK + (long)T_STEPS * FOURR;
    float*  C   = H  + (long)BATCH * RANK;
    float*  Zr  = C  + (long)BATCH * RANK;

    k_init<<<1, 32, 0, stream>>>(acc);

    {   // masked GEMM residual: 16x80 tiles per wave
        int waves  = ((N_SENS + 15) / 16) * (T_STEPS / 80);
        k_main_resid<<<(waves + 7) / 8, 256, 0, stream>>>(W, X, gt, mask, acc);
    }
    k_sumsq<<<256, 256, 0, stream>>>(W, (long)N_SENS * RANK, acc, 1);
    k_sumsq<<<256, 256, 0, stream>>>(X, (long)T_STEPS * RANK, acc, 2);
    k_graph<<<(NEDGE + 255) / 256, 256, 0, stream>>>(W, esrc, edst, acc);

    {   // XK = X @ lstm_kernel
        int waves = (T_STEPS / 16) * (FOURR / 16);
        k_gemm_store<<<(waves + 7) / 8, 256, 0, stream>>>(X, lk, XK, T_STEPS, RANK, FOURR);
    }
    int gateBlocks = (BATCH * RANK + 255) / 256;
    int recWaves   = (BATCH / 16) * (FOURR / 16);
    int recBlocks  = (recWaves + 7) / 8;
    // step 1: h=c=0 -> z = XK[t] + bias
    k_lstm_gate<<<gateBlocks, 256, 0, stream>>>(XK, nullptr, lb, H, C, 0, 1);
    // step 2: z = XK[t+286] + H1 @ rec + bias
    k_gemm_store<<<recBlocks, 256, 0, stream>>>(H, lrk, Zr, BATCH, RANK, FOURR);
    k_lstm_gate<<<gateBlocks, 256, 0, stream>>>(XK, Zr, lb, H, C, 286, 0);
    // step 3: z = XK[t+287] + H2 @ rec + bias
    k_gemm_store<<<recBlocks, 256, 0, stream>>>(H, lrk, Zr, BATCH, RANK, FOURR);
    k_lstm_gate<<<gateBlocks, 256, 0, stream>>>(XK, Zr, lb, H, C, 287, 0);

    {   // dense + squared error vs X[t+288]
        int waves = (BATCH / 16) * 4;
        k_dense_resid<<<(waves + 7) / 8, 256, 0, stream>>>(H, dw, db, X, acc);
    }
    k_final<<<1, 32, 0, stream>>>(acc, (float*)d_out);
}